// RPNLoss_30520037605488
// MI455X (gfx1250) — compile-verified
//
#include <hip/hip_runtime.h>
#include <limits.h>

typedef __attribute__((ext_vector_type(2))) float v2f;
typedef __attribute__((ext_vector_type(8))) float v8f;

#define NH   5
#define NA   9
#define NW   200
#define NHT  200
#define NG   20
#define NANCH (NH*NA*NW*NHT)      // 1,800,000
#define PLANE (NW*NHT)            // 40,000
#define NBLOCKS 1200
#define NTHREADS 256

// ws layout (32-bit words):
//   [0..19]   winner[j] per gt   (atomicMax, init -1)
//   [20]      first kept anchor  (atomicMin, init INT_MAX)
//   [21]      n_keep             (atomicAdd, init 0)
//   [32..32+NBLOCKS) per-block BCE partial sums (float)

__global__ void rpn_init(int* wsi) {
    int t = threadIdx.x;
    if (t < NG)       wsi[t]  = -1;
    else if (t == 20) wsi[20] = INT_MAX;
    else if (t == 21) wsi[21] = 0;
}

__global__ void rpn_main(const float* __restrict__ cls,
                         const float* __restrict__ reg,
                         const float* __restrict__ gt,
                         int* __restrict__ wsi,
                         float* __restrict__ partial)
{
    __shared__ float sgx1[NG], sgy1[NG], sgx2[NG], sgy2[NG], sga[NG];
    __shared__ int   swin[NG];
    __shared__ int   sfirst, snkeep;
    __shared__ float swave[NTHREADS / 32];

    const int tid = threadIdx.x;
    if (tid < NG) {
        float x1 = gt[tid*4+0], y1 = gt[tid*4+1];
        float x2 = gt[tid*4+2], y2 = gt[tid*4+3];
        sgx1[tid] = x1; sgy1[tid] = y1; sgx2[tid] = x2; sgy2[tid] = y2;
        sga[tid]  = (x2 - x1) * (y2 - y1);
        swin[tid] = -1;
    }
    if (tid == 0) { sfirst = INT_MAX; snkeep = 0; }
    __syncthreads();

    int wmax[NG];
#pragma unroll
    for (int j = 0; j < NG; ++j) wmax[j] = -1;

    float bce_acc = 0.0f;
    int   my_first = INT_MAX;
    int   my_keep  = 0;

    const int stride = NBLOCKS * NTHREADS;
    for (int m = blockIdx.x * NTHREADS + tid; m < NANCH; m += stride) {
        // m follows memory order (head, a, w, h) -> coalesced loads
        int head = m / (NA * PLANE);
        int r    = m - head * (NA * PLANE);
        int a    = r / PLANE;
        int r2   = r - a * PLANE;
        int w    = r2 / NHT;
        int h    = r2 - w * NHT;
        // reference flat anchor order (head, w, h, a)
        int n = ((head * NW + w) * NHT + h) * NA + a;

        int rb = ((head * NA + a) * 4) * PLANE + w * NHT + h;
        float bx1 = reg[rb];
        float by1 = reg[rb + PLANE];
        float bx2 = reg[rb + 2 * PLANE];
        float by2 = reg[rb + 3 * PLANE];
        float area_b = (bx2 - bx1) * (by2 - by1);

        bool seen = false, any_pos = false;
        int  wrong = 0;
#pragma unroll
        for (int j = 0; j < NG; ++j) {
            float iw = fmaxf(fminf(bx2, sgx2[j]) - fmaxf(bx1, sgx1[j]), 0.0f);
            float ih = fmaxf(fminf(by2, sgy2[j]) - fmaxf(by1, sgy1[j]), 0.0f);
            float inter = iw * ih;
            float uni   = area_b + sga[j] - inter;
            bool upos = uni > 0.0f;
            bool pos  = upos && (inter >= 0.7f * uni);   // iou >= 0.7
            bool low  = !upos || (inter < 0.3f * uni);   // iou <  0.3
            bool gz   = upos && (inter > 0.0f);          // iou >  0
            if (!seen) {                                  // reach[j]
                if (low) wrong++;
                if (gz && n > wmax[j]) wmax[j] = n;       // candidate winner
            }
            if (pos) { any_pos = true; seen = true; }
        }

        float label = any_pos ? 1.0f : -1.0f;
        if (wrong == NG) label = 0.0f;
        if (label != -1.0f) {                // kept
            my_keep++;
            if (n < my_first) my_first = n;
            float p = cls[m];
            bce_acc += (label == 1.0f) ? -logf(p) : -logf(1.0f - p);
        }
    }

    // flush per-thread reductions into shared (integer atomics: order-independent)
#pragma unroll
    for (int j = 0; j < NG; ++j)
        if (wmax[j] >= 0) atomicMax(&swin[j], wmax[j]);
    if (my_first != INT_MAX) atomicMin(&sfirst, my_first);
    if (my_keep)             atomicAdd(&snkeep, my_keep);

    // --- WMMA wave-level sum of bce_acc ---------------------------------
    // A[16x4] holds each lane's value once (other slot zero); B = ones.
    // D[m,n] = rowsum_m; summing the 8 D VGPRs per lane covers 8 rows,
    // lanes L and L^16 cover complementary rows -> s + shfl_xor(s,16)
    // equals the sum of all 32 lane values (layout-robust).
    v2f av; av.x = bce_acc; av.y = 0.0f;
    v2f bv; bv.x = 1.0f;    bv.y = 1.0f;
    v8f cz = {};
    v8f d = __builtin_amdgcn_wmma_f32_16x16x4_f32(
        false, av, false, bv, (short)0, cz, false, false);
    float s = d[0] + d[1] + d[2] + d[3] + d[4] + d[5] + d[6] + d[7];
    s += __shfl_xor(s, 16, 32);

    int wave = tid >> 5, lane = tid & 31;
    if (lane == 0) swave[wave] = s;
    __syncthreads();

    if (tid == 0) {
        float bsum = 0.0f;
#pragma unroll
        for (int i = 0; i < NTHREADS / 32; ++i) bsum += swave[i];
        partial[blockIdx.x] = bsum;            // fixed slot -> deterministic
        atomicAdd(&wsi[21], snkeep);
        atomicMin(&wsi[20], sfirst);
    }
    if (tid < NG) atomicMax(&wsi[tid], swin[tid]);
}

__global__ void rpn_final(const float* __restrict__ cls,
                          const float* __restrict__ reg,
                          const float* __restrict__ gt,
                          const int* __restrict__ wsi,
                          const float* __restrict__ partial,
                          float* __restrict__ out)
{
    __shared__ double sred[NTHREADS];
    const int tid = threadIdx.x;
    double acc = 0.0;
    for (int i = tid; i < NBLOCKS; i += NTHREADS) acc += (double)partial[i];
    sred[tid] = acc;
    __syncthreads();
    for (int off = NTHREADS / 2; off > 0; off >>= 1) {
        if (tid < off) sred[tid] += sred[tid + off];
        __syncthreads();
    }
    if (tid != 0) return;

    double bce   = sred[0];
    int n_keep   = wsi[21];
    int first    = wsi[20];

    // winner fixups (<= 20 anchors, deduped)
    int done[NG]; int ndone = 0;
    for (int j = 0; j < NG; ++j) {
        int wn = wsi[j];
        if (wn < 0) wn = 0;
        bool dup = false;
        for (int k = 0; k < ndone; ++k) if (done[k] == wn) { dup = true; break; }
        if (dup) continue;
        done[ndone++] = wn;

        int a = wn % NA; int t = wn / NA;
        int h = t % NHT; t /= NHT;
        int w = t % NW;  int head = t / NW;
        int rb = ((head * NA + a) * 4) * PLANE + w * NHT + h;
        float bx1 = reg[rb], by1 = reg[rb + PLANE];
        float bx2 = reg[rb + 2 * PLANE], by2 = reg[rb + 3 * PLANE];
        float area_b = (bx2 - bx1) * (by2 - by1);

        bool seen = false, any_pos = false; int wrong = 0;
        for (int j2 = 0; j2 < NG; ++j2) {
            float gx1 = gt[j2*4+0], gy1 = gt[j2*4+1];
            float gx2 = gt[j2*4+2], gy2 = gt[j2*4+3];
            float iw = fmaxf(fminf(bx2, gx2) - fmaxf(bx1, gx1), 0.0f);
            float ih = fmaxf(fminf(by2, gy2) - fmaxf(by1, gy1), 0.0f);
            float inter = iw * ih;
            float uni = area_b + (gx2 - gx1) * (gy2 - gy1) - inter;
            bool upos = uni > 0.0f;
            if (!seen && (!upos || inter < 0.3f * uni)) wrong++;
            if (upos && inter >= 0.7f * uni) { any_pos = true; seen = true; }
        }
        float label = any_pos ? 1.0f : -1.0f;
        if (wrong == NG) label = 0.0f;

        int cm = (head * NA + a) * PLANE + w * NHT + h;
        float p = cls[cm];
        if (label == -1.0f) {              // newly kept, becomes label 1
            n_keep += 1;
            bce += (double)(-logf(p));
            if (wn < first) first = wn;
        } else if (label == 0.0f) {        // was kept with label 0 -> label 1
            bce += (double)(-logf(p) + logf(1.0f - p));
        }                                  // label 1: no change
    }

    double cls_loss = bce / (double)n_keep;

    // reg loss: first kept anchor's box vs all gt, smooth-L1 mean
    int a = first % NA; int t = first / NA;
    int h = t % NHT; t /= NHT;
    int w = t % NW;  int head = t / NW;
    int rb = ((head * NA + a) * 4) * PLANE + w * NHT + h;
    float b0[4] = { reg[rb], reg[rb + PLANE], reg[rb + 2*PLANE], reg[rb + 3*PLANE] };
    double rsum = 0.0;
    for (int j = 0; j < NG; ++j)
        for (int c = 0; c < 4; ++c) {
            float dd = fabsf(b0[c] - gt[j*4 + c]);
            rsum += (double)((dd < 1.0f) ? 0.5f * dd * dd : dd - 0.5f);
        }
    double reg_loss = rsum / (double)(NG * 4);

    out[0] = (float)(cls_loss / 256.0 + reg_loss / 2400.0);
}

extern "C" void kernel_launch(void* const* d_in, const int* in_sizes, int n_in,
                              void* d_out, int out_size, void* d_ws, size_t ws_size,
                              hipStream_t stream) {
    const float* cls = (const float*)d_in[0];  // [5,1,9,200,200]
    const float* reg = (const float*)d_in[1];  // [5,1,36,200,200]
    const float* gt  = (const float*)d_in[2];  // [20,4]
    int*   wsi     = (int*)d_ws;
    float* partial = (float*)d_ws + 32;
    float* out     = (float*)d_out;

    rpn_init<<<1, 64, 0, stream>>>(wsi);
    rpn_main<<<NBLOCKS, NTHREADS, 0, stream>>>(cls, reg, gt, wsi, partial);
    rpn_final<<<1, NTHREADS, 0, stream>>>(cls, reg, gt, wsi, partial, out);
}